// QuantumTurbulentPDE_8615704396408
// MI455X (gfx1250) — compile-verified
//
#include <hip/hip_runtime.h>
#include <math.h>

// ---------------------------------------------------------------------------
// QuantumTurbulentPDE on MI455X (gfx1250, wave32).
//
// Analytic-derivative formulation of the reference (1 hidden tanh layer):
//   z_i   = sum_j c_j W1[j,i] + b1_i ; h=tanh(z); s=1-h^2; g=-2 h s
//   out_k = sum_i h_i W2[i,k] + b2_k
//   jac[k][j] = sum_i (s_i W1[j,i]) W2[i,k]
//   lap[k]    = sum_i (g_i * (W1[0,i]^2+W1[1,i]^2+W1[2,i]^2)) W2[i,k]
//
// All matmuls via V_WMMA_F32_16X16X4_F32 (K=4 == coordinate dim), transposed
// so that points live on lanes throughout:
//   stage1: z^T  = W1^T_tile (16x4) x c^T (4x16pts)      -> C-layout
//   stage2: acc += W2^T_chunk(16x4) x G_chunk(4x16pts)   (6 accumulators)
// C-layout -> B-layout conversion for stage2 uses 2 ds_bpermutes per chunk.
// tanh is a straight-line v_exp/v_rcp sequence; W2 operand loads use clamped
// addressing + cndmask so the whole tile loop is branch-free (EXEC stays
// all-ones, as WMMA requires).
// ---------------------------------------------------------------------------

typedef __attribute__((ext_vector_type(2))) float v2f;
typedef __attribute__((ext_vector_type(8))) float v8f;

#define HID 256

static __device__ __forceinline__ v8f wmma4(v2f a, v2f b, v8f c) {
  // D = A(16x4 f32) * B(4x16 f32) + C(16x16 f32)
  return __builtin_amdgcn_wmma_f32_16x16x4_f32(
      /*neg_a=*/false, a, /*neg_b=*/false, b,
      /*c_mod=*/(short)0, c, /*reuse_a=*/false, /*reuse_b=*/false);
}

// Branch-free tanh: tanh(x) = sign(x) * (1 - 2/(e^{2|x|}+1)).
// e^{2|x|} = 2^{2*log2(e)*|x|} -> single v_exp_f32; v_rcp_f32 for the divide.
// |x| large -> exp = +inf -> rcp = 0 -> result = +/-1 (correct saturation).
static __device__ __forceinline__ float fast_tanh(float x) {
  const float ax = __builtin_fabsf(x);
  const float e  = __builtin_amdgcn_exp2f(ax * 2.885390081777927f);
  const float r  = 1.0f - 2.0f * __builtin_amdgcn_rcpf(e + 1.0f);
  return __builtin_copysignf(r, x);
}

// Build a stage-2 B operand (rows K0..K0+3 of a 16x16 C-layout tile G,
// all 16 point-columns) in WMMA B-layout:
//   b.x: lanes0-15 -> row K0,   lanes16-31 -> row K0+2
//   b.y: lanes0-15 -> row K0+1, lanes16-31 -> row K0+3
// G (C-layout): reg r holds row r on lanes0-15 and row 8+r on lanes16-31,
// column = lane&15.
template <int K0>
static __device__ __forceinline__ v2f bchunk(const float (&G)[8], int lane) {
  const int nl = lane & 15;
  const bool hi = lane >= 16;
  v2f b;
  if constexpr (K0 < 8) {
    // rows K0..K0+3 live on lanes 0-15 of regs K0..K0+3
    float tx = __shfl(G[K0 + 2], nl, 32);
    float ty = __shfl(G[K0 + 3], nl, 32);
    b.x = hi ? tx : G[K0];
    b.y = hi ? ty : G[K0 + 1];
  } else {
    // rows K0..K0+3 = 8+R0..11+R0 live on lanes 16-31 of regs R0..R0+3
    constexpr int R0 = K0 - 8;
    float tx = __shfl(G[R0 + 0], nl + 16, 32);
    float ty = __shfl(G[R0 + 1], nl + 16, 32);
    b.x = hi ? G[R0 + 2] : tx;
    b.y = hi ? G[R0 + 3] : ty;
  }
  return b;
}

template <int KC>
static __device__ __forceinline__ void stage2(
    const float* __restrict__ w2lane,  // = W2 + min(nl,4), per-lane clamped col
    bool valid,                        // nl < 5 (loop-invariant)
    int i0, int ph, int lane,
    const float (&Gh)[8], const float (&Gj0)[8], const float (&Gj1)[8],
    const float (&Gj2)[8], const float (&Gj3)[8], const float (&Gl)[8],
    v8f& aOut, v8f& aJ0, v8f& aJ1, v8f& aJ2, v8f& aJ3, v8f& aL) {
  constexpr int K0 = KC * 4;
  // A2 = W2^T chunk: M = output index (lane&15, 5 valid), K = 4 hidden units
  //   a.x: K = i0+K0+2*ph ; a.y: K = i0+K0+2*ph+1   (A-layout)
  // Unconditional clamped loads (always in-bounds), then zero invalid cols.
  const int rowa = i0 + K0 + 2 * ph;
  const float vx = w2lane[rowa * 5];
  const float vy = w2lane[(rowa + 1) * 5];
  v2f a2;
  a2.x = valid ? vx : 0.0f;
  a2.y = valid ? vy : 0.0f;
  aOut = wmma4(a2, bchunk<K0>(Gh, lane), aOut);
  aJ0  = wmma4(a2, bchunk<K0>(Gj0, lane), aJ0);
  aJ1  = wmma4(a2, bchunk<K0>(Gj1, lane), aJ1);
  aJ2  = wmma4(a2, bchunk<K0>(Gj2, lane), aJ2);
  aJ3  = wmma4(a2, bchunk<K0>(Gj3, lane), aJ3);
  aL   = wmma4(a2, bchunk<K0>(Gl, lane), aL);
}

__global__ __launch_bounds__(256) void qtpde_wmma_kernel(
    const float* __restrict__ coords, const float* __restrict__ W1,
    const float* __restrict__ b1, const float* __restrict__ W2,
    const float* __restrict__ b2, const float* __restrict__ hbarp,
    const float* __restrict__ viscp, const float* __restrict__ coupp,
    float* __restrict__ out, int npts) {
  const int tid  = blockIdx.x * blockDim.x + threadIdx.x;
  const int lane = threadIdx.x & 31;
  const int wave = tid >> 5;
  const int pt0  = wave * 16;            // 16 points per wave
  if (pt0 >= npts) return;               // wave-uniform

  const int nl = lane & 15;
  const int ph = lane >> 4;              // lane half

  // Loop-invariant, per-lane clamped W2 column pointer + validity mask.
  const bool valid = (nl < 5);
  const float* __restrict__ w2lane = W2 + (valid ? nl : 4);

  // B (stage 1) = c^T (4x16 points), constant across tiles.
  //   b.x: lanes0-15 -> c[pt][0], lanes16-31 -> c[pt][2]; b.y -> c[pt][1]/[3]
  v2f bC;
  {
    const int p = (pt0 + nl < npts) ? (pt0 + nl) : (npts - 1);
    const float* cp = coords + (size_t)p * 4 + 2 * ph;
    bC.x = cp[0];
    bC.y = cp[1];
  }

  v8f aOut = {}, aJ0 = {}, aJ1 = {}, aJ2 = {}, aJ3 = {}, aL = {};

  for (int t = 0; t < HID / 16; ++t) {
    const int i0 = t * 16;

    // A (stage 1) = W1^T tile (16 hid x 4): A-layout
    //   a.x = W1[2*ph+0][i0+nl], a.y = W1[2*ph+1][i0+nl]
    v2f aW1;
    aW1.x = W1[(2 * ph + 0) * HID + i0 + nl];
    aW1.y = W1[(2 * ph + 1) * HID + i0 + nl];

    // C init = b1 broadcast over point columns; row = i0 + r + 8*ph
    v8f zc;
#pragma unroll
    for (int r = 0; r < 8; ++r) zc[r] = b1[i0 + r + 8 * ph];

    // z^T tile (C-layout: hid rows on regs/halves, points on lanes)
    v8f z = wmma4(aW1, bC, zc);

    float Gh[8], Gj0[8], Gj1[8], Gj2[8], Gj3[8], Gl[8];
#pragma unroll
    for (int r = 0; r < 8; ++r) {
      const int m = r + 8 * ph;  // local hidden row of this register
      // Redistribute W1 tile from A-layout lanes to this row (per-lane bcast)
      const float w0 = __shfl(aW1.x, m, 32);       // W1[0][i0+m]
      const float w1 = __shfl(aW1.y, m, 32);       // W1[1][i0+m]
      const float w2 = __shfl(aW1.x, m + 16, 32);  // W1[2][i0+m]
      const float w3 = __shfl(aW1.y, m + 16, 32);  // W1[3][i0+m]
      const float h = fast_tanh(z[r]);
      const float s = 1.0f - h * h;
      const float g = -2.0f * h * s;
      Gh[r]  = h;
      Gj0[r] = s * w0;
      Gj1[r] = s * w1;
      Gj2[r] = s * w2;
      Gj3[r] = s * w3;
      Gl[r]  = g * (w0 * w0 + w1 * w1 + w2 * w2);
    }

    stage2<0>(w2lane, valid, i0, ph, lane, Gh, Gj0, Gj1, Gj2, Gj3, Gl, aOut, aJ0, aJ1, aJ2, aJ3, aL);
    stage2<1>(w2lane, valid, i0, ph, lane, Gh, Gj0, Gj1, Gj2, Gj3, Gl, aOut, aJ0, aJ1, aJ2, aJ3, aL);
    stage2<2>(w2lane, valid, i0, ph, lane, Gh, Gj0, Gj1, Gj2, Gj3, Gl, aOut, aJ0, aJ1, aJ2, aJ3, aL);
    stage2<3>(w2lane, valid, i0, ph, lane, Gh, Gj0, Gj1, Gj2, Gj3, Gl, aOut, aJ0, aJ1, aJ2, aJ3, aL);
  }

  // Epilogue: lanes 0-15 each own one point; accumulator reg k = output k.
  if (lane < 16 && (pt0 + lane) < npts) {
    const int p = pt0 + lane;
    const float hbar = hbarp[0];
    const float visc = viscp[0];
    const float coup = coupp[0];

    float o[5], lap[5], jac[5][4];
#pragma unroll
    for (int k = 0; k < 5; ++k) {
      o[k]      = aOut[k] + b2[k];
      lap[k]    = aL[k];
      jac[k][0] = aJ0[k];
      jac[k][1] = aJ1[k];
      jac[k][2] = aJ2[k];
      jac[k][3] = aJ3[k];
    }

    const float psi_r = o[0], psi_i = o[1];
    const float u[3] = {o[2], o[3], o[4]};
    const float q_r = -hbar * jac[1][3] + 0.5f * hbar * hbar * lap[0];
    const float q_i =  hbar * jac[0][3] + 0.5f * hbar * hbar * lap[1];
    const float rho = psi_r * psi_r + psi_i * psi_i;
    float Jv[3];
#pragma unroll
    for (int b = 0; b < 3; ++b) Jv[b] = psi_r * jac[0][b] + psi_i * jac[1][b];

    float ns[3];
    float div = 0.0f;
#pragma unroll
    for (int a = 0; a < 3; ++a) {
      const float conv =
          jac[2 + a][0] * u[0] + jac[2 + a][1] * u[1] + jac[2 + a][2] * u[2];
      ns[a] = jac[2 + a][3] + conv - visc * lap[2 + a] -
              coup * (Jv[a] - u[a] * rho);
      div += jac[2 + a][0] + jac[2 + a][1] + jac[2 + a][2] + jac[2 + a][3];
    }

    float* op = out + (size_t)p * 10;
    op[0] = q_r;
    op[1] = q_i;
    op[2] = ns[0];
    op[3] = ns[1];
    op[4] = ns[2];
    op[5] = div;
    op[6] = rho;
    op[7] = Jv[0];
    op[8] = Jv[1];
    op[9] = Jv[2];
  }
}

extern "C" void kernel_launch(void* const* d_in, const int* in_sizes, int n_in,
                              void* d_out, int out_size, void* d_ws,
                              size_t ws_size, hipStream_t stream) {
  const float* coords = (const float*)d_in[0];
  const float* W1     = (const float*)d_in[1];
  const float* b1     = (const float*)d_in[2];
  const float* W2     = (const float*)d_in[3];
  const float* b2     = (const float*)d_in[4];
  const float* hbar   = (const float*)d_in[5];
  const float* visc   = (const float*)d_in[6];
  const float* coup   = (const float*)d_in[7];
  float* out = (float*)d_out;

  const int npts     = in_sizes[0] / 4;
  const int nwaves   = (npts + 15) / 16;
  const int nthreads = nwaves * 32;
  dim3 block(256);
  dim3 grid((nthreads + 255) / 256);
  qtpde_wmma_kernel<<<grid, block, 0, stream>>>(coords, W1, b1, W2, b2, hbar,
                                                visc, coup, out, npts);
}